// Osses2021_4148938408290
// MI455X (gfx1250) — compile-verified
//
#include <hip/hip_runtime.h>
#include <hip/hip_bf16.h>
#include <math.h>

// ---------------------------------------------------------------------------
// Osses2021 auditory model for MI455X (gfx1250, wave32).
// Stage 0/1 (FIR cascade) -> fused single conv done as f32 WMMA GEMM,
//   filter pre-packed in WMMA A-fragment order for coalesced b64 loads,
//   x window stored time-reversed in LDS so B fragments are ascending
//   adjacent words (ds_load_2addr returns them in fragment order, no movs).
// Stage 2/3 (IHC + nonlinear adaptation) -> one resident block, serial in T,
//   fast transcendentals (v_rcp_f32 / v_exp_f32) to shorten the serial chain.
// Stage 4 (modulation filterbank) -> 2976-thread kernel, serial in T.
// ---------------------------------------------------------------------------

typedef __attribute__((ext_vector_type(2))) float v2f;
typedef __attribute__((ext_vector_type(4))) float v4f;
typedef __attribute__((ext_vector_type(8))) float v8f;

constexpr int   B_    = 8;
constexpr int   T_    = 22050;
constexpr int   NCH   = 31;
constexpr int   LG    = 2048;
constexpr int   LH    = 512;
constexpr int   M_    = 12;
constexpr float FSF   = 44100.0f;
constexpr float PIF   = 3.14159265358979f;

constexpr int   LHM   = LH + LH - 1;       // hp (x) me length      = 1023
constexpr int   LCOMB = LHM + LG - 1;      // full cascade length   = 3070
constexpr int   KPAD  = 3072;              // K rounded to mult of 4
constexpr int   KT4   = KPAD / 4;          // 768 K-tiles
constexpr int   NPAD  = 32;                // channel dim padded
constexpr int   TPAD  = ((T_ + 63) / 64) * 64;   // 22080 (guard-free stores)

// workspace layout (float offsets)
constexpr size_t WS_HM = 0;                                   // 1024
constexpr size_t WS_HA = 1024;                                // 2*KT4*32*2 = 98304
constexpr size_t WS_YT = WS_HA + (size_t)2 * KT4 * 32 * 2;    // TPAD*256
constexpr size_t WS_AD = WS_YT + (size_t)TPAD * 256;          // T*256

// ---------------------------------------------------------------------------
// Kernel B1: hm = hp (x) me   (1023 taps)
// ---------------------------------------------------------------------------
__global__ void build_hm_k(const float* __restrict__ hp,
                           const float* __restrict__ me,
                           float* __restrict__ hm) {
    for (int i = threadIdx.x; i < LHM; i += blockDim.x) {
        int j0 = i - (LH - 1); if (j0 < 0) j0 = 0;
        int j1 = (i < LH - 1) ? i : (LH - 1);
        float s = 0.0f;
        for (int j = j0; j <= j1; ++j) s += hp[j] * me[i - j];
        hm[i] = s;
    }
}

// ---------------------------------------------------------------------------
// Kernel B2: build combined filter directly in WMMA A-fragment order.
//   hA[((ntile*KT4 + kb4)*32 + lane)] = float2{ A[m][k], A[m][k+1] }
//   where m = lane&15, k = kb4*4 + (lane>>4)*2, A[n][k] = (hm (x) gt[n])[k].
// ---------------------------------------------------------------------------
__device__ __forceinline__ float hcomb_val(const float* __restrict__ hm,
                                           const float* __restrict__ gt,
                                           int n, int k) {
    if (n >= NCH || k >= LCOMB) return 0.0f;
    int j0 = k - (LG - 1); if (j0 < 0) j0 = 0;
    int j1 = (k < LHM - 1) ? k : (LHM - 1);
    const float* g = gt + (size_t)n * LG;
    float s = 0.0f;
    for (int j = j0; j <= j1; ++j) s += hm[j] * g[k - j];
    return s;
}

__global__ void build_hA_k(const float* __restrict__ hm,
                           const float* __restrict__ gt,
                           float* __restrict__ hA) {
    int idx = blockIdx.x * blockDim.x + threadIdx.x;   // 0 .. 2*KT4*32-1
    int lane  = idx & 31;
    int kb4   = (idx >> 5) % KT4;
    int ntile = (idx >> 5) / KT4;
    int m    = lane & 15;
    int koff = (lane >> 4) * 2;
    int n    = ntile * 16 + m;
    int k    = kb4 * 4 + koff;
    hA[2 * (size_t)idx]     = hcomb_val(hm, gt, n, k);
    hA[2 * (size_t)idx + 1] = hcomb_val(hm, gt, n, k + 1);
}

// ---------------------------------------------------------------------------
// Kernel A: causal conv as f32 WMMA GEMM.
//   y[b,n,t] = sum_k h[n,k] * x[b, t-k]
// Block = 128 threads (4 waves). Each wave: one 16(n) x 16(t) tile.
// Block covers 64 consecutive t, sharing the (time-reversed) x window in LDS.
// A fragments: one coalesced b64 load per WMMA. B fragments: ascending
// adjacent LDS pair (ds_load_2addr in fragment order).
// Output time-major (padded, guard-free b128 stores): yT[t*256 + b*32 + n].
// ---------------------------------------------------------------------------
__global__ __launch_bounds__(128) void fir_wmma_k(const float* __restrict__ x,
                                                  const float* __restrict__ hA,
                                                  float* __restrict__ yT) {
    // reversed window: xs[i] = x[b, tblk + 63 - i], i in [0, 63 + LCOMB]
    constexpr int XSZ = 64 + LCOMB + 2;    // 3136
    __shared__ float xs[XSZ];

    const int tid  = threadIdx.x;
    const int wave = tid >> 5;
    const int lane = tid & 31;

    const int tblk  = blockIdx.x * 64;
    const int ntile = blockIdx.y;          // 0..1
    const int bidx  = blockIdx.z;          // 0..7

    for (int i = tid; i < XSZ; i += 128) {
        int g = tblk + 63 - i;
        xs[i] = (g >= 0 && g < T_) ? x[(size_t)bidx * T_ + g] : 0.0f;
    }
    __syncthreads();

    const int col  = lane & 15;            // B/D column (N) == A row (M)
    const int koff = (lane >> 4) * 2;      // K pair selected by lane half
    // xs index of B[kk][col] = x[t0+col-kk] is (63 - wave*16 - col + kk)
    const int pbase = 63 - wave * 16 - col + koff;

    // pre-packed A fragments for this ntile, this lane
    const v2f* __restrict__ hAv =
        (const v2f*)hA + (size_t)ntile * KT4 * 32 + lane;

    v8f c = {0.f, 0.f, 0.f, 0.f, 0.f, 0.f, 0.f, 0.f};

#pragma unroll 8
    for (int kb4 = 0; kb4 < KT4; ++kb4) {
        v2f a = hAv[(size_t)kb4 * 32];                 // coalesced b64
        const int p = pbase + kb4 * 4;
        v2f bfrag;
        bfrag.x = xs[p];                               // B[kk][col]
        bfrag.y = xs[p + 1];                           // B[kk+1][col]
        c = __builtin_amdgcn_wmma_f32_16x16x4_f32(
                /*neg_a=*/false, a, /*neg_b=*/false, bfrag,
                /*c_mod=*/(short)0, c, /*reuse_a=*/false, /*reuse_b=*/false);
    }

    // D: VGPR r -> rows r (lanes 0-15) / r+8 (lanes 16-31); rows are
    // consecutive channels in yT -> two b128 stores, no guards (padded dims).
    const int rbase = (lane >> 4) * 8;
    const int t = tblk + wave * 16 + col;
    float* dst = yT + (size_t)t * 256 + bidx * NPAD + ntile * 16 + rbase;
    v4f lo = {c[0], c[1], c[2], c[3]};
    v4f hi = {c[4], c[5], c[6], c[7]};
    *(v4f*)dst       = lo;
    *(v4f*)(dst + 4) = hi;
}

// ---------------------------------------------------------------------------
// Kernel C: IHC envelope (5 cascaded one-pole LP @2kHz on halfwave-rectified
// signal) + 5 nonlinear adaptation loops. One thread per (b,n) channel,
// single resident block, all state in registers, serial over T.
// ---------------------------------------------------------------------------
__global__ __launch_bounds__(256, 1) void ihc_adapt_k(const float* __restrict__ yT,
                                                      float* __restrict__ ad) {
    const int tid = threadIdx.x;           // b = tid>>5, n = tid&31 (n==31 idle)

    const float a_lp = __expf(-2.0f * PIF * 2000.0f / FSF);
    const float b_lp = 1.0f - a_lp;

    const float taus[5] = {0.005f, 0.05f, 0.129f, 0.253f, 0.5f};
    float a1[5], b0a[5], st[5], fac[5], efc[5], off[5];
#pragma unroll
    for (int j = 0; j < 5; ++j) {
        a1[j]  = expf(-1.0f / (FSF * taus[j]));
        b0a[j] = 1.0f - a1[j];
        float ini = powf(1e-5f, exp2f(-(float)(j + 1)));
        st[j]  = ini;
        float mv = (1.0f - ini * ini) * 5.0f - 1.0f;
        fac[j] = 2.0f * mv;
        efc[j] = -2.0f / mv;
        off[j] = mv - 1.0f;
    }
    const float corr  = powf(1e-5f, 0.03125f);     // 1e-5 ^ 2^-5
    const float scale = 100.0f / (1.0f - corr);

    float s0 = 0.f, s1 = 0.f, s2 = 0.f, s3 = 0.f, s4 = 0.f;

    for (int t = 0; t < T_; ++t) {
        float hw = fmaxf(yT[(size_t)t * 256 + tid], 0.0f);
        s0 = a_lp * s0 + b_lp * hw;
        s1 = a_lp * s1 + b_lp * s0;
        s2 = a_lp * s2 + b_lp * s1;
        s3 = a_lp * s3 + b_lp * s2;
        s4 = a_lp * s4 + b_lp * s3;

        float tmp = fmaxf(s4, 1e-5f);
#pragma unroll
        for (int j = 0; j < 5; ++j) {
            tmp = tmp * __builtin_amdgcn_rcpf(st[j]);          // v_rcp_f32
            if (tmp > 1.0f) {
                float e = __expf(efc[j] * (tmp - 1.0f));       // v_exp_f32 path
                tmp = fac[j] * __builtin_amdgcn_rcpf(1.0f + e) - off[j];
            }
            st[j] = a1[j] * st[j] + b0a[j] * tmp;
        }
        ad[(size_t)t * 256 + tid] = (tmp - corr) * scale;
    }
}

// ---------------------------------------------------------------------------
// Kernel D: modulation filterbank — one complex one-pole resonator per thread
// (b,n,m), serial over T, writes final [B,N,M,T] output.
// ---------------------------------------------------------------------------
__global__ void modfb_k(const float* __restrict__ ad, float* __restrict__ out) {
    const int idx = blockIdx.x * blockDim.x + threadIdx.x;
    if (idx >= B_ * NCH * M_) return;
    const int m  = idx % M_;
    const int bn = idx / M_;
    const int b  = bn / NCH;
    const int n  = bn % NCH;

    const float mfc_t[12] = {2.5f, 5.0f, 10.0f, 16.7f, 27.8f, 46.3f,
                             77.2f, 128.6f, 214.3f, 357.2f, 595.4f, 992.3f};
    const float mfc = mfc_t[m];
    const float r   = expf(-PIF * (mfc * 0.5f) / FSF);
    const float w   = 2.0f * PIF * mfc / FSF;
    const float pr  = r * cosf(w);
    const float pim = r * sinf(w);
    const float b0  = 1.0f - r;            // b0m is real
    const bool  low = (mfc <= 10.0f);
    const float att = 0.70710678f;

    float sr = 0.f, si = 0.f;
    float* o = out + ((size_t)(b * NCH + n) * M_ + m) * T_;
    const float* a = ad + (size_t)b * NPAD + n;

    for (int t = 0; t < T_; ++t) {
        float xv = a[(size_t)t * 256];
        float nr = pr * sr - pim * si + b0 * xv;
        float ni = pr * si + pim * sr;
        sr = nr; si = ni;
        o[t] = low ? sr
                   : att * __builtin_amdgcn_sqrtf(sr * sr + si * si + 1e-12f);
    }
}

// ---------------------------------------------------------------------------
extern "C" void kernel_launch(void* const* d_in, const int* in_sizes, int n_in,
                              void* d_out, int out_size, void* d_ws, size_t ws_size,
                              hipStream_t stream) {
    const float* x  = (const float*)d_in[0];
    const float* hp = (const float*)d_in[1];
    const float* me = (const float*)d_in[2];
    const float* gt = (const float*)d_in[3];
    float* out = (float*)d_out;
    float* ws  = (float*)d_ws;

    float* hm = ws + WS_HM;
    float* hA = ws + WS_HA;
    float* yT = ws + WS_YT;
    float* ad = ws + WS_AD;

    // fused filter construction (pre-packed in WMMA A-fragment order)
    build_hm_k<<<1, 256, 0, stream>>>(hp, me, hm);
    build_hA_k<<<(2 * KT4 * 32) / 256, 256, 0, stream>>>(hm, gt, hA);

    // WMMA FIR cascade: grid = (t-tiles of 64, n-tiles of 16, batch)
    dim3 gA(TPAD / 64, 2, B_);
    fir_wmma_k<<<gA, 128, 0, stream>>>(x, hA, yT);

    // serial recurrences
    ihc_adapt_k<<<1, 256, 0, stream>>>(yT, ad);
    modfb_k<<<(B_ * NCH * M_ + 127) / 128, 128, 0, stream>>>(ad, out);
}